// shGLM_32770600469058
// MI455X (gfx1250) — compile-verified
//
#include <hip/hip_runtime.h>
#include <hip/hip_bf16.h>

#define T_LEN   20000
#define E_NUM   2000
#define SUB_NO  15
#define G_NUM   4
#define W_SIZE  201
#define HIST_T  50
#define NCOL    64      // 15*4 = 60 active columns, padded to 64

typedef __attribute__((ext_vector_type(2))) float v2f;
typedef __attribute__((ext_vector_type(8))) float v8f;

// workspace layout in floats
#define OFF_PT   0                           // 64 x 2000 (transposed projection)
#define OFF_KG   (OFF_PT + NCOL * E_NUM)     // 4 x 201 group kernels (W folded in)
#define OFF_FK   (OFF_KG + 1024)             // 15 x 50 history kernels
#define OFF_Z    (OFF_FK + 1024)             // 20000 x 64 GEMM output
#define OFF_SIN  (OFF_Z + T_LEN * NCOL)      // 15 x 20000 subunit inputs
#define OFF_OUT  (OFF_SIN + SUB_NO * T_LEN)  // 15 x 20000 subunit outputs

// ---------------------------------------------------------------------------
// Prep: synaptic group kernels (W folded) + history basis kernels
// ---------------------------------------------------------------------------
__global__ void k_prep_small(const float* __restrict__ Delta,
                             const float* __restrict__ W,
                             const float* __restrict__ Tau,
                             const float* __restrict__ K_hist,
                             float* __restrict__ ws) {
    int idx = blockIdx.x * blockDim.x + threadIdx.x;
    float* Kg = ws + OFF_KG;
    float* FK = ws + OFF_FK;
    if (idx < G_NUM * W_SIZE) {
        int g = idx / W_SIZE, k = idx % W_SIZE;
        float t = (float)k - __expf(Delta[0]);
        t = fmaxf(t, 0.f);
        float tauf = __expf(Tau[g]);
        float taus = tauf * 2.8f + 10.4f;
        float tf = t / tauf, ts = t / taus;
        float kern = (ts * __expf(-ts) * 0.3f + tf * __expf(-tf)) / 1.3f;
        Kg[idx] = W[g] * kern;
    } else if (idx < G_NUM * W_SIZE + SUB_NO * HIST_T) {
        int j = idx - G_NUM * W_SIZE;
        int s = j / HIST_T, h = j % HIST_T;
        const float tau3[3] = {2.f, 8.f, 22.f};
        float acc = 0.f;
        #pragma unroll
        for (int b = 0; b < 3; ++b) {
            float tt = (float)h / tau3[b];
            acc += K_hist[s * 3 + b] * tt * __expf(-tt);
        }
        FK[s * HIST_T + h] = acc;
    }
}

// ---------------------------------------------------------------------------
// Prep: transposed projection PT[c][e] = mask[s][e] * (G_syn[e]==g), c=4s+g
// ---------------------------------------------------------------------------
__global__ void k_prep_P(const int* __restrict__ G_syn,
                         const int* __restrict__ mask,
                         float* __restrict__ ws) {
    int idx = blockIdx.x * blockDim.x + threadIdx.x;
    if (idx >= NCOL * E_NUM) return;
    int c = idx / E_NUM, e = idx % E_NUM;
    float v = 0.f;
    if (c < SUB_NO * G_NUM) {
        int g = c & 3, s = c >> 2;
        v = (G_syn[e] == g && mask[s * E_NUM + e] != 0) ? 1.f : 0.f;
    }
    ws[OFF_PT + idx] = v;
}

// ---------------------------------------------------------------------------
// GEMM: Z[t, c] = sum_e X[t,e] * PT[c,e]  via V_WMMA_F32_16X16X4_F32.
// One wave per 16-row M tile; 4 N-tiles (64 cols) accumulated per wave.
// fp32 WMMA keeps full precision; loop is HBM-bound on X (160MB @ 23.3TB/s).
// ---------------------------------------------------------------------------
__global__ void k_gemm_wmma(const float* __restrict__ X,
                            float* __restrict__ ws) {
    const float* PT = ws + OFF_PT;
    float* Z = ws + OFF_Z;
    int lane = threadIdx.x & 31;
    int wave = blockIdx.x * (blockDim.x >> 5) + (threadIdx.x >> 5);
    int t0 = wave * 16;
    if (t0 >= T_LEN) return;          // wave-uniform: EXEC all-1 for WMMA

    int m    = lane & 15;             // A: row (M);  B: column (N)
    int koff = (lane >> 4) << 1;      // lanes 16-31 hold K=2,3

    const float* xr = X  + (size_t)(t0 + m) * E_NUM + koff;
    const float* p0 = PT + (size_t)(m +  0) * E_NUM + koff;
    const float* p1 = PT + (size_t)(m + 16) * E_NUM + koff;
    const float* p2 = PT + (size_t)(m + 32) * E_NUM + koff;
    const float* p3 = PT + (size_t)(m + 48) * E_NUM + koff;

    v8f acc0 = {0,0,0,0,0,0,0,0};
    v8f acc1 = acc0, acc2 = acc0, acc3 = acc0;

    #pragma unroll 4
    for (int k = 0; k < E_NUM; k += 4) {
        __builtin_prefetch(xr + k + 256, 0, 0);   // global_prefetch_b8
        v2f a  = *(const v2f*)(xr + k);           // A[m][k..k+1] (per half-wave K split)
        v2f b0 = *(const v2f*)(p0 + k);           // B[k..k+1][n], n = lane&15
        v2f b1 = *(const v2f*)(p1 + k);
        v2f b2 = *(const v2f*)(p2 + k);
        v2f b3 = *(const v2f*)(p3 + k);
        acc0 = __builtin_amdgcn_wmma_f32_16x16x4_f32(false, a, false, b0, (short)0, acc0, false, false);
        acc1 = __builtin_amdgcn_wmma_f32_16x16x4_f32(false, a, false, b1, (short)0, acc1, false, false);
        acc2 = __builtin_amdgcn_wmma_f32_16x16x4_f32(false, a, false, b2, (short)0, acc2, false, false);
        acc3 = __builtin_amdgcn_wmma_f32_16x16x4_f32(false, a, false, b3, (short)0, acc3, false, false);
    }

    // D layout: VGPR r -> M = r + 8*(lane>=16), N = lane&15
    int mrow = (lane >> 4) << 3;
    int n = lane & 15;
    #pragma unroll
    for (int r = 0; r < 8; ++r) {
        size_t row = (size_t)(t0 + mrow + r) * NCOL;
        Z[row + n +  0] = acc0[r];
        Z[row + n + 16] = acc1[r];
        Z[row + n + 32] = acc2[r];
        Z[row + n + 48] = acc3[r];
    }
}

// ---------------------------------------------------------------------------
// Temporal correlation: subunit_in[s][t] = sum_g sum_k Kg[g][k]*Z[t+k-100, 4s+g]
// (zero-padded). One thread per time step, all 15 subunits in registers.
// ---------------------------------------------------------------------------
__global__ void k_conv_time(float* __restrict__ ws) {
    const float* Z  = ws + OFF_Z;
    const float* Kg = ws + OFF_KG;
    float* SIN = ws + OFF_SIN;
    int t = blockIdx.x * blockDim.x + threadIdx.x;
    if (t >= T_LEN) return;
    float acc[SUB_NO];
    #pragma unroll
    for (int s = 0; s < SUB_NO; ++s) acc[s] = 0.f;
    int klo = (t < 100) ? (100 - t) : 0;
    int khi = (t > T_LEN - 101) ? (T_LEN + 100 - t) : W_SIZE;
    for (int k = klo; k < khi; ++k) {
        const float* zr = Z + (size_t)(t + k - 100) * NCOL;
        float k0 = Kg[k];
        float k1 = Kg[W_SIZE + k];
        float k2 = Kg[2 * W_SIZE + k];
        float k3 = Kg[3 * W_SIZE + k];
        #pragma unroll
        for (int s = 0; s < SUB_NO; ++s)
            acc[s] += k0 * zr[4*s] + k1 * zr[4*s+1] + k2 * zr[4*s+2] + k3 * zr[4*s+3];
    }
    #pragma unroll
    for (int s = 0; s < SUB_NO; ++s) SIN[(size_t)s * T_LEN + t] = acc[s];
}

// ---------------------------------------------------------------------------
// Sequential subunit scan: one wave32 per subunit. Lane l holds history
// sp[t-1-l] (a) and sp[t-33-l] (b); 50-tap feedback = 2 FMAs + shfl_xor
// reduction; history shift via shfl_up. Launched level-by-level (leaves->root).
// ---------------------------------------------------------------------------
__global__ void k_scan(float* __restrict__ ws,
                       const float* __restrict__ Theta,
                       const float* __restrict__ C,
                       const float* __restrict__ spike,
                       const float* __restrict__ Vo,
                       float* __restrict__ d_out,
                       int s_first) {
    int s = s_first + blockIdx.x;
    int lane = threadIdx.x;
    const float* SIN = ws + OFF_SIN;
    const float* FK  = ws + OFF_FK;
    float* OUT = ws + OFF_OUT;

    float wa = FK[s * HIST_T + lane];
    float wb = (lane < HIST_T - 32) ? FK[s * HIST_T + 32 + lane] : 0.f;
    float th = Theta[s];
    float ec = __expf(C[s]);
    float sz = spike[s];

    const float* in = SIN + (size_t)s * T_LEN;
    int c1 = 2 * s + 1, c2 = 2 * s + 2;
    const float* ch1 = (c1 < SUB_NO) ? (OUT + (size_t)c1 * T_LEN) : nullptr;
    const float* ch2 = (c2 < SUB_NO) ? (OUT + (size_t)c2 * T_LEN) : nullptr;
    float* o  = (s == 0) ? d_out : (OUT + (size_t)s * T_LEN);
    float vo  = (s == 0) ? Vo[0] : 0.f;

    float a = 0.f, b = 0.f;   // distributed history registers
    for (int t = 0; t < T_LEN; ++t) {
        float tot = in[t];
        if (ch1) tot += ch1[t];
        if (ch2) tot += ch2[t];
        float base = 1.f / (1.f + __expf(th - tot));     // sigmoid(tot - Theta)
        float partial = wa * a + wb * b;
        #pragma unroll
        for (int off = 16; off > 0; off >>= 1)
            partial += __shfl_xor(partial, off, 32);
        float thr = base + partial;
        float sp  = 1.f / (1.f + __expf(-thr));
        if (lane == 0) o[t] = (thr + sp * sz) * ec + vo;
        float a31 = __shfl(a, 31, 32);
        a = __shfl_up(a, 1, 32);
        b = __shfl_up(b, 1, 32);
        if (lane == 0) { a = sp; b = a31; }
    }
}

// ---------------------------------------------------------------------------
extern "C" void kernel_launch(void* const* d_in, const int* in_sizes, int n_in,
                              void* d_out, int out_size, void* d_ws, size_t ws_size,
                              hipStream_t stream) {
    const float* X      = (const float*)d_in[0];
    const float* V_o    = (const float*)d_in[1];
    const float* Delta  = (const float*)d_in[2];
    const float* W      = (const float*)d_in[3];
    const float* Tau    = (const float*)d_in[4];
    const float* C      = (const float*)d_in[5];
    const float* Theta  = (const float*)d_in[6];
    const float* spike  = (const float*)d_in[7];
    const float* K_hist = (const float*)d_in[8];
    const int*   G_syn  = (const int*)d_in[9];
    const int*   mask   = (const int*)d_in[10];
    float* out = (float*)d_out;
    float* ws  = (float*)d_ws;

    // small kernels (1554 elements)
    k_prep_small<<<7, 256, 0, stream>>>(Delta, W, Tau, K_hist, ws);
    // projection matrix 64 x 2000
    k_prep_P<<<(NCOL * E_NUM + 255) / 256, 256, 0, stream>>>(G_syn, mask, ws);
    // WMMA GEMM: 1250 waves of 16 rows, 8 waves per block
    k_gemm_wmma<<<(T_LEN / 16 + 7) / 8, 256, 0, stream>>>(X, ws);
    // temporal correlation
    k_conv_time<<<(T_LEN + 255) / 256, 256, 0, stream>>>(ws);
    // tree scan, leaves -> root (levels are serialized on the stream)
    k_scan<<<8, 32, 0, stream>>>(ws, Theta, C, spike, V_o, out, 7);
    k_scan<<<4, 32, 0, stream>>>(ws, Theta, C, spike, V_o, out, 3);
    k_scan<<<2, 32, 0, stream>>>(ws, Theta, C, spike, V_o, out, 1);
    k_scan<<<1, 32, 0, stream>>>(ws, Theta, C, spike, V_o, out, 0);
}